// GNNStack_59785944760753
// MI455X (gfx1250) — compile-verified
//
#include <hip/hip_runtime.h>
#include <hip/hip_bf16.h>
#include <hip/amd_detail/amd_hip_unsafe_atomics.h>

// ---------------------------------------------------------------------------
// GAT stack for MI455X (gfx1250). fp32 WMMA (V_WMMA_F32_16X16X4_F32) for all
// dense GEMMs (exact fp32 semantics; GEMMs are not the bottleneck — edge
// scatter/gather is). wave32 everywhere.
// ---------------------------------------------------------------------------

typedef __attribute__((ext_vector_type(2))) float v2f;
typedef __attribute__((ext_vector_type(8))) float v8f;

#define IN_DIM 128
#define HID 64
#define OUT_DIM 47
#define NEG_SLOPE 0.2f
#define EPS_DENOM 1e-16f

// ---- order-preserving float<->uint encoding for atomic segment-max --------
__device__ __forceinline__ unsigned fenc(float f) {
  unsigned u = __float_as_uint(f);
  return u ^ ((u & 0x80000000u) ? 0xFFFFFFFFu : 0x80000000u);
}
__device__ __forceinline__ float fdec(unsigned k) {
  unsigned u = k ^ ((k & 0x80000000u) ? 0x80000000u : 0xFFFFFFFFu);
  return __uint_as_float(u);
}

// ---------------------------------------------------------------------------
// GEMM: Y[M x 64] = act(X[M x K]) @ W[K x 64] + bias.   M % 32 == 0.
// Block: 256 threads = 8 waves; block tile 32 rows x 64 cols; wave tile 16x16.
// WMMA f32 16x16x4 fragment layout (ISA 7.12.2):
//   A 16x4:  lane<16 -> row=lane,    a0=K+0,a1=K+1 ; lane>=16 -> a0=K+2,a1=K+3
//   B 4x16:  vgpr0: row K+0 (lanes 0-15) / K+2 (lanes 16-31); vgpr1: K+1/K+3
//   C/D:     vgpr r: lane<16 -> (M=r, N=lane) ; lane>=16 -> (M=r+8, N=lane-16)
// ---------------------------------------------------------------------------
template <int K>
__global__ __launch_bounds__(256) void k_gemm(const float* __restrict__ X,
                                              const float* __restrict__ W,
                                              const float* __restrict__ bias,
                                              float* __restrict__ Y,
                                              int relu_in) {
  constexpr int LDX = K + 4;                 // pad to avoid LDS bank conflicts
  __shared__ float sx[32 * LDX];
  __shared__ float sw[K * 64];

  const int tid = threadIdx.x;
  const int row_base = blockIdx.x * 32;

  // stage X tile (32 x K), with optional input ReLU
  for (int idx = tid; idx < 32 * K; idx += 256) {
    int r = idx / K, c = idx - r * K;
    float v = X[(size_t)(row_base + r) * K + c];
    if (relu_in) v = v > 0.f ? v : 0.f;
    sx[r * LDX + c] = v;
  }
  // stage W (K x 64)
  for (int idx = tid; idx < K * 64; idx += 256) sw[idx] = W[idx];
  __syncthreads();

  const int wave = tid >> 5;
  const int lane = tid & 31;
  const int half = lane >> 4;      // 0: lanes 0-15, 1: lanes 16-31
  const int l15  = lane & 15;
  const int trow = (wave >> 2) * 16;   // 0 or 16
  const int col0 = (wave & 3) * 16;    // 0,16,32,48

  v8f acc = {};
#pragma unroll
  for (int k = 0; k < K; k += 4) {
    v2f a, b;
    const float* ap = &sx[(trow + l15) * LDX + k + 2 * half];
    a[0] = ap[0];
    a[1] = ap[1];
    b[0] = sw[(k + 2 * half + 0) * 64 + col0 + l15];
    b[1] = sw[(k + 2 * half + 1) * 64 + col0 + l15];
    acc = __builtin_amdgcn_wmma_f32_16x16x4_f32(false, a, false, b,
                                                (short)0, acc, false, false);
  }

#pragma unroll
  for (int r = 0; r < 8; ++r) {
    int row = row_base + trow + r + half * 8;
    int col = col0 + l15;
    Y[(size_t)row * 64 + col] = acc[r] + bias[col];
  }
}

// ---------------------------------------------------------------------------
// Per-node attention coefficients: al[n]=<h[n],att_l>, ar[n]=<h[n],att_r>.
// One wave per node (HID=64 -> 2 elems/lane), shuffle reduction.
// ---------------------------------------------------------------------------
__global__ __launch_bounds__(256) void k_alpha(const float* __restrict__ h,
                                               const float* __restrict__ attl,
                                               const float* __restrict__ attr,
                                               float* __restrict__ al,
                                               float* __restrict__ ar, int n) {
  int node = (int)((blockIdx.x * blockDim.x + threadIdx.x) >> 5);
  int lane = threadIdx.x & 31;
  if (node >= n) return;
  const float* row = h + (size_t)node * 64;
  float h0 = row[lane], h1 = row[lane + 32];
  float pl = h0 * attl[lane] + h1 * attl[lane + 32];
  float pr = h0 * attr[lane] + h1 * attr[lane + 32];
#pragma unroll
  for (int off = 16; off > 0; off >>= 1) {
    pl += __shfl_xor(pl, off, 32);
    pr += __shfl_xor(pr, off, 32);
  }
  if (lane == 0) {
    al[node] = pl;
    ar[node] = pr;
  }
}

// ---------------------------------------------------------------------------
// Edge pass 1: e = leaky_relu(al[src]+ar[dst]); store e; segment-max via
// order-preserving uint atomicMax (mkey pre-zeroed == encoded minimum).
// ---------------------------------------------------------------------------
__global__ __launch_bounds__(256) void k_edge_max(const int* __restrict__ src,
                                                  const int* __restrict__ dst,
                                                  const float* __restrict__ al,
                                                  const float* __restrict__ ar,
                                                  float* __restrict__ ebuf,
                                                  unsigned* __restrict__ mkey,
                                                  int E) {
  int e = blockIdx.x * blockDim.x + threadIdx.x;
  if (e >= E) return;
  int s = src[e], d = dst[e];
  float v = al[s] + ar[d];
  v = v > 0.f ? v : NEG_SLOPE * v;
  ebuf[e] = v;
  atomicMax(&mkey[d], fenc(v));
}

// ---------------------------------------------------------------------------
// Edge pass 2: ex = exp(e - m[dst]); store ex; denom[dst] += ex.
// ---------------------------------------------------------------------------
__global__ __launch_bounds__(256) void k_edge_exp(const int* __restrict__ dst,
                                                  float* __restrict__ ebuf,
                                                  const unsigned* __restrict__ mkey,
                                                  float* __restrict__ denom,
                                                  int E) {
  int e = blockIdx.x * blockDim.x + threadIdx.x;
  if (e >= E) return;
  int d = dst[e];
  float m = fdec(mkey[d]);
  float ex = __expf(ebuf[e] - m);
  ebuf[e] = ex;
  unsafeAtomicAdd(&denom[d], ex);
}

// ---------------------------------------------------------------------------
// Edge pass 3: out[dst] += (ex / (denom[dst]+eps)) * h[src].
// One wave per edge; each lane handles a float2 of the 64-wide channel.
// ---------------------------------------------------------------------------
__global__ __launch_bounds__(256) void k_edge_agg(const int* __restrict__ src,
                                                  const int* __restrict__ dst,
                                                  const float* __restrict__ ebuf,
                                                  const float* __restrict__ denom,
                                                  const float* __restrict__ h,
                                                  float* __restrict__ out,
                                                  int E) {
  int e = (int)((blockIdx.x * blockDim.x + threadIdx.x) >> 5);
  int lane = threadIdx.x & 31;
  if (e >= E) return;
  int s = src[e], d = dst[e];
  float a = ebuf[e] / (denom[d] + EPS_DENOM);
  int c = lane * 2;
  const float2 hv = *(const float2*)&h[(size_t)s * 64 + c];
  float* op = &out[(size_t)d * 64 + c];
  unsafeAtomicAdd(op + 0, a * hv.x);
  unsafeAtomicAdd(op + 1, a * hv.y);
}

// ---------------------------------------------------------------------------
// Final: logits = t @ Wp2 + bp2 (64x47), then log_softmax over 47.
// One thread per node; t row is 256B contiguous; Wp2 (12KB) lives in L2/WGP$.
// ---------------------------------------------------------------------------
__global__ __launch_bounds__(256) void k_post(const float* __restrict__ t,
                                              const float* __restrict__ W,
                                              const float* __restrict__ b,
                                              float* __restrict__ out, int n) {
  int i = blockIdx.x * blockDim.x + threadIdx.x;
  if (i >= n) return;
  float acc[OUT_DIM];
#pragma unroll
  for (int j = 0; j < OUT_DIM; ++j) acc[j] = b[j];
  const float* row = t + (size_t)i * 64;
#pragma unroll 4
  for (int c = 0; c < 64; ++c) {
    float xv = row[c];
    const float* wr = &W[c * OUT_DIM];
#pragma unroll
    for (int j = 0; j < OUT_DIM; ++j) acc[j] += xv * wr[j];
  }
  float m = acc[0];
#pragma unroll
  for (int j = 1; j < OUT_DIM; ++j) m = fmaxf(m, acc[j]);
  float s = 0.f;
#pragma unroll
  for (int j = 0; j < OUT_DIM; ++j) s += __expf(acc[j] - m);
  float lse = m + __logf(s);
  float* op = &out[(size_t)i * OUT_DIM];
#pragma unroll
  for (int j = 0; j < OUT_DIM; ++j) op[j] = acc[j] - lse;
}

// ---------------------------------------------------------------------------
extern "C" void kernel_launch(void* const* d_in, const int* in_sizes, int n_in,
                              void* d_out, int out_size, void* d_ws,
                              size_t ws_size, hipStream_t stream) {
  const float* x = (const float*)d_in[0];
  const int* ei = (const int*)d_in[1];  // jax default x64-off -> int32
  const int N = in_sizes[0] / IN_DIM;
  const int E = in_sizes[1] / 2;
  const int* src = ei;
  const int* dst = ei + E;

  const float* W[3]  = {(const float*)d_in[2], (const float*)d_in[6],  (const float*)d_in[10]};
  const float* B[3]  = {(const float*)d_in[3], (const float*)d_in[7],  (const float*)d_in[11]};
  const float* AL[3] = {(const float*)d_in[4], (const float*)d_in[8],  (const float*)d_in[12]};
  const float* AR[3] = {(const float*)d_in[5], (const float*)d_in[9],  (const float*)d_in[13]};
  const float* Wp1 = (const float*)d_in[14];
  const float* bp1 = (const float*)d_in[15];
  const float* Wp2 = (const float*)d_in[16];
  const float* bp2 = (const float*)d_in[17];

  // workspace carve-out (256B aligned slices)
  char* ws = (char*)d_ws;
  size_t off = 0;
  auto carve = [&](size_t bytes) -> void* {
    void* p = ws + off;
    off += (bytes + 255) & ~(size_t)255;
    return p;
  };
  float* bufA      = (float*)carve((size_t)N * 64 * 4);  // h (linear output)
  float* bufB      = (float*)carve((size_t)N * 64 * 4);  // aggregated output
  float* al        = (float*)carve((size_t)N * 4);
  float* ar        = (float*)carve((size_t)N * 4);
  unsigned* mkey   = (unsigned*)carve((size_t)N * 4);
  float* denom     = (float*)carve((size_t)N * 4);
  float* ebuf      = (float*)carve((size_t)E * 4);

  const int gemm_blocks  = N / 32;            // N % 32 == 0 (100000/32 = 3125)
  const int alpha_blocks = (N * 32 + 255) / 256;
  const int edge_blocks  = (E + 255) / 256;
  const int agg_blocks   = (E * 32 + 255) / 256;

  for (int i = 0; i < 3; ++i) {
    const float* in = (i == 0) ? x : bufB;
    if (i == 0)
      k_gemm<IN_DIM><<<gemm_blocks, 256, 0, stream>>>(in, W[0], B[0], bufA, 0);
    else
      k_gemm<HID><<<gemm_blocks, 256, 0, stream>>>(in, W[i], B[i], bufA, 1);

    k_alpha<<<alpha_blocks, 256, 0, stream>>>(bufA, AL[i], AR[i], al, ar, N);

    hipMemsetAsync(mkey, 0, (size_t)N * 4, stream);   // encoded minimum
    hipMemsetAsync(denom, 0, (size_t)N * 4, stream);

    k_edge_max<<<edge_blocks, 256, 0, stream>>>(src, dst, al, ar, ebuf, mkey, E);
    k_edge_exp<<<edge_blocks, 256, 0, stream>>>(dst, ebuf, mkey, denom, E);

    hipMemsetAsync(bufB, 0, (size_t)N * 64 * 4, stream);
    k_edge_agg<<<agg_blocks, 256, 0, stream>>>(src, dst, ebuf, denom, bufA,
                                               bufB, E);
  }

  // post MLP: bufA = relu(bufB) @ Wp1 + bp1 ; then 64x47 + log_softmax
  k_gemm<HID><<<gemm_blocks, 256, 0, stream>>>(bufB, Wp1, bp1, bufA, 1);
  k_post<<<(N + 255) / 256, 256, 0, stream>>>(bufA, Wp2, bp2, (float*)d_out, N);
}